// GLIFTorch_57226144252673
// MI455X (gfx1250) — compile-verified
//
#include <hip/hip_runtime.h>
#include <hip/hip_bf16.h>

#define T_STEPS 10

typedef float v4f __attribute__((ext_vector_type(4)));

// ---------------------------------------------------------------------------
// CDNA5 async global->LDS copy (ASYNCcnt-tracked), 16 bytes per lane.
// Inline asm per CDNA5 ISA: the clang builtin exists but
// expects CUDA-LangAS-qualified int4 pointers that HIP source cannot spell
// reliably, so we emit the instruction directly.
//   global_load_async_to_lds_b128 vdst(LDS byte offset), vaddr(64b), off
// Low 32 bits of a flat shared-aperture address == LDS byte offset (ISA 10.2).
// ---------------------------------------------------------------------------
__device__ __forceinline__ void async_copy16(void* lds_dst, const float* gsrc) {
  unsigned int lds_off = (unsigned int)(unsigned long long)lds_dst;
  unsigned long long gaddr = (unsigned long long)gsrc;
  asm volatile("global_load_async_to_lds_b128 %0, %1, off"
               :: "v"(lds_off), "v"(gaddr)
               : "memory");
}

__device__ __forceinline__ void wait_async_all() {
#if __has_builtin(__builtin_amdgcn_s_wait_asynccnt)
  __builtin_amdgcn_s_wait_asynccnt(0);
#else
  asm volatile("s_wait_asynccnt 0" ::: "memory");
#endif
  asm volatile("" ::: "memory");  // compiler ordering fence for the LDS reads
}

__device__ __forceinline__ float sigf(float v) {
  return 1.0f / (1.0f + expf(-v));
}

// ---------------------------------------------------------------------------
// Pre-pass: fold all scalar params into per-timestep coefficients (uniform
// across the 8.4M elements). cf layout: [t*5 + {A,B,C,D,Vth_s}], cf[5*T] = ga.
//   u' = A*u*(1 - ga*o) + B + C*x - D*o ;  o' = (u' > Vth_s)
// ---------------------------------------------------------------------------
__global__ void glif_coeff_kernel(const float* __restrict__ alpha,
                                  const float* __restrict__ beta,
                                  const float* __restrict__ gamma,
                                  const float* __restrict__ tau,
                                  const float* __restrict__ Vth,
                                  const float* __restrict__ leak,
                                  const float* __restrict__ conduct,
                                  const float* __restrict__ reVth,
                                  float* __restrict__ cf) {
  const int t = threadIdx.x;
  const float al = (alpha[0] > 0.0f) ? 1.0f : 0.0f;
  const float be = (beta[0]  > 0.0f) ? 1.0f : 0.0f;
  const float ga = (gamma[0] > 0.0f) ? 1.0f : 0.0f;
  if (t == 0) cf[5 * T_STEPS] = ga;
  if (t < T_STEPS) {
    const float tau_s   = sigf(tau[t]);
    const float Vth_s   = sigf(Vth[t]);
    const float leak_s  = sigf(leak[t]);
    const float cond_s  = sigf(conduct[t]);
    const float reVth_s = sigf(reVth[t]);
    cf[t * 5 + 0] = 1.0f - al * (1.0f - tau_s);      // A: u multiplier
    cf[t * 5 + 1] = -(1.0f - al) * leak_s;           // B: additive leak
    cf[t * 5 + 2] = 1.0f - be * (1.0f - cond_s);     // C: x multiplier
    cf[t * 5 + 3] = (1.0f - ga) * reVth_s;           // D: reset term
    cf[t * 5 + 4] = Vth_s;                           // threshold
  }
}

// ---------------------------------------------------------------------------
// Main scan: each lane owns 4 consecutive floats for all 10 timesteps.
// Stage x via 10 outstanding async B128 copies into this lane's LDS slots,
// wait ASYNCcnt==0, run the recurrence in registers, NT-store the spikes.
// LDS: 10 * 256 * 16B = 40 KB/block -> 8 blocks per 320KB WGP.
// ---------------------------------------------------------------------------
__global__ __launch_bounds__(256) void glif_scan_kernel(
    const float* __restrict__ x, const float* __restrict__ cf,
    float* __restrict__ out, int N) {
  __shared__ v4f tile[T_STEPS][256];
  const int tid = threadIdx.x;
  const long long base = ((long long)blockIdx.x * 256 + tid) * 4;
  if (base >= N) return;

#pragma unroll
  for (int t = 0; t < T_STEPS; ++t) {
    async_copy16(&tile[t][tid], x + (long long)t * N + base);
  }
  wait_async_all();

  const float ga = cf[5 * T_STEPS];
  float u0 = 0.f, u1 = 0.f, u2 = 0.f, u3 = 0.f;
  float o0 = 0.f, o1 = 0.f, o2 = 0.f, o3 = 0.f;

#pragma unroll
  for (int t = 0; t < T_STEPS; ++t) {
    const float A = cf[t * 5 + 0];
    const float B = cf[t * 5 + 1];
    const float C = cf[t * 5 + 2];
    const float D = cf[t * 5 + 3];
    const float V = cf[t * 5 + 4];
    const v4f xv = tile[t][tid];

    const float n0 = A * u0 * (1.0f - ga * o0) + B + C * xv.x - D * o0;
    const float n1 = A * u1 * (1.0f - ga * o1) + B + C * xv.y - D * o1;
    const float n2 = A * u2 * (1.0f - ga * o2) + B + C * xv.z - D * o2;
    const float n3 = A * u3 * (1.0f - ga * o3) + B + C * xv.w - D * o3;
    o0 = (n0 > V) ? 1.0f : 0.0f;  u0 = n0;
    o1 = (n1 > V) ? 1.0f : 0.0f;  u1 = n1;
    o2 = (n2 > V) ? 1.0f : 0.0f;  u2 = n2;
    o3 = (n3 > V) ? 1.0f : 0.0f;  u3 = n3;

    v4f ov;
    ov.x = o0; ov.y = o1; ov.z = o2; ov.w = o3;
    __builtin_nontemporal_store(ov, (v4f*)(out + (long long)t * N + base));
  }
}

extern "C" void kernel_launch(void* const* d_in, const int* in_sizes, int n_in,
                              void* d_out, int out_size, void* d_ws, size_t ws_size,
                              hipStream_t stream) {
  const float* x       = (const float*)d_in[0];
  const float* alpha   = (const float*)d_in[1];
  const float* beta    = (const float*)d_in[2];
  const float* gamma   = (const float*)d_in[3];
  const float* tau     = (const float*)d_in[4];
  const float* Vth     = (const float*)d_in[5];
  const float* leak    = (const float*)d_in[6];
  const float* conduct = (const float*)d_in[7];
  const float* reVth   = (const float*)d_in[8];
  float* out = (float*)d_out;
  float* cf  = (float*)d_ws;  // 51 floats of scratch

  const int N = in_sizes[0] / T_STEPS;  // elements per timestep (8,388,608)

  glif_coeff_kernel<<<1, 32, 0, stream>>>(alpha, beta, gamma, tau, Vth, leak,
                                          conduct, reVth, cf);

  const int nvec   = (N + 3) / 4;            // float4 lanes
  const int blocks = (nvec + 255) / 256;     // 8192 for the reference shape
  glif_scan_kernel<<<blocks, 256, 0, stream>>>(x, cf, out, N);
}